// JacobiConv_41120016892645
// MI455X (gfx1250) — compile-verified
//
#include <hip/hip_runtime.h>
#include <hip/hip_bf16.h>

// ---------------------------------------------------------------------------
// JacobiConv for MI455X (gfx1250, wave32).
//  * Encoder GEMMs use V_WMMA_F32_16X16X4_F32 (exact fp32, matches reference).
//  * SpMM is edge-parallel with global_atomic_add_f32 (L2-resident accumulator).
// ---------------------------------------------------------------------------

typedef float v2f __attribute__((ext_vector_type(2)));
typedef float v8f __attribute__((ext_vector_type(8)));

#define LDX 520   // LDS row stride for x tile  (512 + pad, even -> b64 loads ok)
#define LDT 264   // LDS row stride for t tile  (256 + pad)

// One wave (32 threads) computes h[r0..r0+15][0..63] = relu(x@W1+b1)@W2 + b2.
__global__ __launch_bounds__(32)
void encoder_kernel(const float* __restrict__ x,
                    const float* __restrict__ W1, const float* __restrict__ b1,
                    const float* __restrict__ W2, const float* __restrict__ b2,
                    float* __restrict__ h, int n)
{
    __shared__ float sx[16 * LDX];   // 16 x 512 input tile
    __shared__ float st[16 * LDT];   // 16 x 256 hidden tile

    const int lane  = threadIdx.x;       // 0..31
    const int row   = lane & 15;         // A-frag row / B,D column within tile
    const int khalf = lane >> 4;         // 0 or 1 -> K pair select
    const int r0    = blockIdx.x * 16;

    // --- stage x tile into LDS (float4 cooperative loads) -------------------
    for (int i = lane; i < 16 * 128; i += 32) {
        int rr = i >> 7;              // row 0..15
        int cc = (i & 127) << 2;      // col 0..508 step 4
        int gr = r0 + rr; if (gr >= n) gr = n - 1;   // clamp (N % 16 == 0 normally)
        const float4 v = *reinterpret_cast<const float4*>(x + (size_t)gr * 512 + cc);
        float* dst = &sx[rr * LDX + cc];
        dst[0] = v.x; dst[1] = v.y; dst[2] = v.z; dst[3] = v.w;
    }
    __syncthreads();

    // --- stage 1: t = relu(x @ W1 + b1), 16 column tiles, K = 512 -----------
    for (int nt = 0; nt < 16; ++nt) {
        const float bias = b1[nt * 16 + row];
        v8f acc;
        #pragma unroll
        for (int i = 0; i < 8; ++i) acc[i] = bias;   // bias broadcast per column

        const float* wcol = W1 + nt * 16 + row;      // column (N) base, row stride 256
        #pragma unroll 4
        for (int k0 = 0; k0 < 512; k0 += 4) {
            const int ka = k0 + 2 * khalf;
            const v2f a = *reinterpret_cast<const v2f*>(&sx[row * LDX + ka]);
            v2f b;
            b.x = wcol[(size_t)ka * 256];
            b.y = wcol[(size_t)(ka + 1) * 256];
            acc = __builtin_amdgcn_wmma_f32_16x16x4_f32(
                      false, a, false, b, (short)0, acc, false, false);
        }
        // ReLU and scatter D frag to LDS: lane holds [M = r + 8*khalf][N = nt*16+row]
        #pragma unroll
        for (int r = 0; r < 8; ++r) {
            float v = acc[r] > 0.f ? acc[r] : 0.f;
            st[(r + 8 * khalf) * LDT + nt * 16 + row] = v;
        }
    }
    __syncthreads();

    // --- stage 2: h = t @ W2 + b2, 4 column tiles, K = 256 ------------------
    for (int nt = 0; nt < 4; ++nt) {
        const float bias = b2[nt * 16 + row];
        v8f acc;
        #pragma unroll
        for (int i = 0; i < 8; ++i) acc[i] = bias;

        const float* wcol = W2 + nt * 16 + row;      // row stride 64
        #pragma unroll 4
        for (int k0 = 0; k0 < 256; k0 += 4) {
            const int ka = k0 + 2 * khalf;
            const v2f a = *reinterpret_cast<const v2f*>(&st[row * LDT + ka]);
            v2f b;
            b.x = wcol[(size_t)ka * 64];
            b.y = wcol[(size_t)(ka + 1) * 64];
            acc = __builtin_amdgcn_wmma_f32_16x16x4_f32(
                      false, a, false, b, (short)0, acc, false, false);
        }
        #pragma unroll
        for (int r = 0; r < 8; ++r) {
            const int m = r0 + r + 8 * khalf;
            if (m < n) h[(size_t)m * 64 + nt * 16 + row] = acc[r];
        }
    }
}

// ---------------------------------------------------------------------------
__global__ void zero_kernel(float* __restrict__ p, int total4)
{
    float4 z4 = make_float4(0.f, 0.f, 0.f, 0.f);
    for (int i = blockIdx.x * blockDim.x + threadIdx.x; i < total4;
         i += gridDim.x * blockDim.x)
        reinterpret_cast<float4*>(p)[i] = z4;
}

// S[dst] += val * P[src] over all edges; 16 lanes per edge, float4 per lane.
__global__ void spmm_kernel(const float* __restrict__ P,
                            const int* __restrict__ esrc,
                            const int* __restrict__ edst,
                            const float* __restrict__ eval,
                            float* __restrict__ S, int E)
{
    const long long stride = (long long)gridDim.x * blockDim.x;
    const long long total  = (long long)E * 16;
    for (long long gt = (long long)blockIdx.x * blockDim.x + threadIdx.x;
         gt < total; gt += stride) {
        const int e = (int)(gt >> 4);
        const int f = ((int)gt & 15) << 2;
        const int s = esrc[e];
        const int d = edst[e];
        const float v = eval[e];
        const float4 p = *reinterpret_cast<const float4*>(P + (size_t)s * 64 + f);
        float* o = S + (size_t)d * 64 + f;
        atomicAdd(o + 0, v * p.x);
        atomicAdd(o + 1, v * p.y);
        atomicAdd(o + 2, v * p.z);
        atomicAdd(o + 3, v * p.w);
    }
}

// k = 1 step:  Pc = c0*h + c1*S ;  z = g0*h + g1*Pc
__global__ void combine1_kernel(const float* __restrict__ h,
                                const float* __restrict__ S,
                                const float* __restrict__ g0,
                                const float* __restrict__ g1,
                                float c0, float c1,
                                float* __restrict__ Pc, float* __restrict__ z,
                                int total4)
{
    for (int i = blockIdx.x * blockDim.x + threadIdx.x; i < total4;
         i += gridDim.x * blockDim.x) {
        const int f = (i << 2) & 63;
        const float4 hv = reinterpret_cast<const float4*>(h)[i];
        const float4 sv = reinterpret_cast<const float4*>(S)[i];
        float4 pc, zv;
        pc.x = c0 * hv.x + c1 * sv.x;  pc.y = c0 * hv.y + c1 * sv.y;
        pc.z = c0 * hv.z + c1 * sv.z;  pc.w = c0 * hv.w + c1 * sv.w;
        zv.x = g0[f + 0] * hv.x + g1[f + 0] * pc.x;
        zv.y = g0[f + 1] * hv.y + g1[f + 1] * pc.y;
        zv.z = g0[f + 2] * hv.z + g1[f + 2] * pc.z;
        zv.w = g0[f + 3] * hv.w + g1[f + 3] * pc.w;
        reinterpret_cast<float4*>(Pc)[i] = pc;
        reinterpret_cast<float4*>(z)[i]  = zv;
    }
}

// k >= 2 step: Pn = th*S + tp*Pc - td*Pp ;  z += gk*Pn
__global__ void combineK_kernel(const float* __restrict__ S,
                                const float* __restrict__ Pc,
                                const float* __restrict__ Pp,
                                const float* __restrict__ gk,
                                float th, float tp, float td,
                                float* __restrict__ Pn, float* __restrict__ z,
                                int total4)
{
    for (int i = blockIdx.x * blockDim.x + threadIdx.x; i < total4;
         i += gridDim.x * blockDim.x) {
        const int f = (i << 2) & 63;
        const float4 sv  = reinterpret_cast<const float4*>(S)[i];
        const float4 pcv = reinterpret_cast<const float4*>(Pc)[i];
        const float4 ppv = reinterpret_cast<const float4*>(Pp)[i];
        float4 pn, zv = reinterpret_cast<float4*>(z)[i];
        pn.x = th * sv.x + tp * pcv.x - td * ppv.x;
        pn.y = th * sv.y + tp * pcv.y - td * ppv.y;
        pn.z = th * sv.z + tp * pcv.z - td * ppv.z;
        pn.w = th * sv.w + tp * pcv.w - td * ppv.w;
        zv.x += gk[f + 0] * pn.x;  zv.y += gk[f + 1] * pn.y;
        zv.z += gk[f + 2] * pn.z;  zv.w += gk[f + 3] * pn.w;
        reinterpret_cast<float4*>(Pn)[i] = pn;
        reinterpret_cast<float4*>(z)[i]  = zv;
    }
}

// ---------------------------------------------------------------------------
extern "C" void kernel_launch(void* const* d_in, const int* in_sizes, int n_in,
                              void* d_out, int out_size, void* d_ws, size_t ws_size,
                              hipStream_t stream)
{
    const float* x     = (const float*)d_in[0];
    const int*   esrc  = (const int*)  d_in[1];
    const int*   edst  = (const int*)  d_in[2];
    const float* eval  = (const float*)d_in[3];
    const float* W1    = (const float*)d_in[4];
    const float* b1    = (const float*)d_in[5];
    const float* W2    = (const float*)d_in[6];
    const float* b2    = (const float*)d_in[7];
    const float* gamma = (const float*)d_in[8];   // [K+1][64]

    const int N = in_sizes[0] / 512;
    const int E = in_sizes[1];
    const int K = 10;
    const double a = 1.0, b = 1.0;

    const size_t cnt = (size_t)N * 64;            // elements per P matrix
    const int total4 = (int)(cnt / 4);

    float* ws = (float*)d_ws;                     // needs 4*cnt floats (~102 MB)
    float* S  = ws;                               // SpMM accumulator
    float* B0 = ws + cnt;                         // rotating P buffers
    float* B1 = ws + 2 * cnt;
    float* B2 = ws + 3 * cnt;

    float* z = (float*)d_out;

    const int eb = 256;
    const int encBlocks  = (N + 15) / 16;
    const int elemBlocks = (total4 + eb - 1) / eb;
    const int spmmBlocks = 8192;

    // h -> B0
    encoder_kernel<<<encBlocks, 32, 0, stream>>>(x, W1, b1, W2, b2, B0, N);

    // k = 1
    zero_kernel<<<elemBlocks, eb, 0, stream>>>(S, total4);
    spmm_kernel<<<spmmBlocks, eb, 0, stream>>>(B0, esrc, edst, eval, S, E);
    const float c0 = (float)((a - b) / 2.0);
    const float c1 = (float)((a + b + 2.0) / 2.0);
    combine1_kernel<<<elemBlocks, eb, 0, stream>>>(B0, S, gamma + 0 * 64,
                                                   gamma + 1 * 64, c0, c1, B1, z,
                                                   total4);

    float *Pp = B0, *Pc = B1, *Pn = B2;
    for (int k = 2; k <= K; ++k) {
        const double kk = (double)k;
        const double th = (2*kk + a + b) * (2*kk + a + b - 1.0) /
                          (2.0 * kk * (kk + a + b));
        const double tp = (2*kk + a + b - 1.0) * (a*a - b*b) /
                          (2.0 * kk * (kk + a + b) * (2*kk + a + b - 2.0));
        const double td = (kk + a - 1.0) * (kk + b - 1.0) * (2*kk + a + b) /
                          (kk * (kk + a + b) * (2*kk + a + b - 2.0));

        zero_kernel<<<elemBlocks, eb, 0, stream>>>(S, total4);
        spmm_kernel<<<spmmBlocks, eb, 0, stream>>>(Pc, esrc, edst, eval, S, E);
        combineK_kernel<<<elemBlocks, eb, 0, stream>>>(S, Pc, Pp, gamma + k * 64,
                                                       (float)th, (float)tp,
                                                       (float)td, Pn, z, total4);
        float* t = Pp; Pp = Pc; Pc = Pn; Pn = t;   // rotate buffers
    }
}